// TopKAttentionPoolingMultiModalProjector_87540023427682
// MI455X (gfx1250) — compile-verified
//
#include <hip/hip_runtime.h>
#include <hip/hip_bf16.h>

// Shapes fixed by the reference
#define S_DIM 576
#define E_DIM 1152
#define T_DIM 4096
#define TOPK  128

typedef __attribute__((ext_vector_type(2))) float v2f;
typedef __attribute__((ext_vector_type(8))) float v8f;

// ---------------------------------------------------------------------------
// Kernel 1: proj[s,t] = sum_e A[s,e] * W[t,e] + bias[t] + pos[s,t]
// Block: 256 threads = 8 waves. Block tile: 64 (M) x 128 (N).
// Each wave computes FOUR 16x16 M-tiles sharing one B fragment per K-step:
//   - B (W_proj) is the dominant memory stream (18.9 MB, reread per M-block);
//     4x M-blocking cuts its L2 traffic from ~680 MB to ~170 MB.
//   - 4 independent accumulator chains let WMMAs issue back-to-back with no
//     D->A/B hazards between them.
// ---------------------------------------------------------------------------
__global__ __launch_bounds__(256) void
proj_gemm_kernel(const float* __restrict__ A,    // [S, E]
                 const float* __restrict__ W,    // [T, E]
                 const float* __restrict__ bias, // [T]
                 const float* __restrict__ pos,  // [S, T]
                 float* __restrict__ proj)       // [S, T]
{
    const int wave = threadIdx.x >> 5;   // 0..7
    const int lane = threadIdx.x & 31;

    const int m0 = blockIdx.y * 64;              // row base (S), 4 tiles of 16
    const int n0 = blockIdx.x * 128 + wave * 16; // col base (T)

    const int half = lane >> 4;   // 0: lanes 0-15, 1: lanes 16-31
    const int sub  = lane & 15;
    const int kOff = half * 2;    // A/B frag: upper lane-half holds K=2,3

    // A fragment source for tile i: row (m0 + i*16 + sub), K pair at kOff.
    // Tile stride 16*E_DIM floats folds into the 24-bit instruction offset.
    const float* aPtr = A + (size_t)(m0 + sub) * E_DIM + kOff;
    // B fragment source: B[k][n] = W[n*E + k]; N = sub, same K pairing
    const float* bPtr = W + (size_t)(n0 + sub) * E_DIM + kOff;

    v8f c0 = {}, c1 = {}, c2 = {}, c3 = {};
    #pragma unroll 2
    for (int k = 0; k < E_DIM; k += 4) {
        const v2f b  = *(const v2f*)(bPtr + k);
        const v2f a0 = *(const v2f*)(aPtr + k);
        const v2f a1 = *(const v2f*)(aPtr + k + 16 * E_DIM);
        const v2f a2 = *(const v2f*)(aPtr + k + 32 * E_DIM);
        const v2f a3 = *(const v2f*)(aPtr + k + 48 * E_DIM);
        c0 = __builtin_amdgcn_wmma_f32_16x16x4_f32(false, a0, false, b, (short)0, c0, false, false);
        c1 = __builtin_amdgcn_wmma_f32_16x16x4_f32(false, a1, false, b, (short)0, c1, false, false);
        c2 = __builtin_amdgcn_wmma_f32_16x16x4_f32(false, a2, false, b, (short)0, c2, false, false);
        c3 = __builtin_amdgcn_wmma_f32_16x16x4_f32(false, a3, false, b, (short)0, c3, false, false);
    }

    // C/D layout: VGPR r -> lanes 0-15: (M=r, N=lane); lanes 16-31: (M=r+8, N=lane-16)
    const int nOut = n0 + sub;
    const float bv = bias[nOut];

    v8f acc[4] = { c0, c1, c2, c3 };
    #pragma unroll
    for (int i = 0; i < 4; ++i) {
        const int mBase = m0 + i * 16 + half * 8;
        #pragma unroll
        for (int r = 0; r < 8; ++r) {
            const int m = mBase + r;
            proj[(size_t)m * T_DIM + nOut] =
                acc[i][r] + bv + pos[(size_t)m * T_DIM + nOut];
        }
    }
}

// ---------------------------------------------------------------------------
// Kernel 2: logits[s] = dot(proj[s,:], Wscore) + bscore   (one block per row)
// ---------------------------------------------------------------------------
__global__ __launch_bounds__(256) void
score_kernel(const float* __restrict__ proj,   // [S, T]
             const float* __restrict__ Wsc,    // [T]
             const float* __restrict__ bsc,    // [1]
             float* __restrict__ logits)       // [S]
{
    __shared__ float red[8];
    const int s = blockIdx.x;
    float acc = 0.0f;
    for (int t = threadIdx.x; t < T_DIM; t += 256)
        acc += proj[(size_t)s * T_DIM + t] * Wsc[t];
    // wave32 reduction
    #pragma unroll
    for (int off = 16; off > 0; off >>= 1)
        acc += __shfl_down(acc, off, 32);
    if ((threadIdx.x & 31) == 0) red[threadIdx.x >> 5] = acc;
    __syncthreads();
    if (threadIdx.x == 0) {
        float t = bsc[0];
        #pragma unroll
        for (int i = 0; i < 8; ++i) t += red[i];
        logits[s] = t;
    }
}

// ---------------------------------------------------------------------------
// Kernel 3: top-128 by logit (softmax is monotonic -> skip it), emit the
// surviving row indices compacted in ascending original order.
// Single 576-thread block; O(S^2) ranking with jax.top_k tie-break (low idx
// wins ties), then position = count of kept indices below s.
// ---------------------------------------------------------------------------
__global__ __launch_bounds__(576) void
topk_kernel(const float* __restrict__ logits, int* __restrict__ map)
{
    __shared__ float l[S_DIM];
    __shared__ int   rank[S_DIM];
    const int s = threadIdx.x;
    l[s] = logits[s];
    __syncthreads();

    const float mine = l[s];
    int r = 0;
    for (int j = 0; j < S_DIM; ++j) {
        const float v = l[j];
        r += (v > mine) || (v == mine && j < s);
    }
    rank[s] = r;
    __syncthreads();

    if (r < TOPK) {
        int pos = 0;
        for (int j = 0; j < s; ++j) pos += (rank[j] < TOPK);
        map[pos] = s;
    }
}

// ---------------------------------------------------------------------------
// Kernel 4: out[k, :] = proj[map[k], :]   (float4 copies)
// grid = (T/1024, 128), block = 256
// ---------------------------------------------------------------------------
__global__ __launch_bounds__(256) void
gather_kernel(const float* __restrict__ proj, const int* __restrict__ map,
              float* __restrict__ out)
{
    const int k = blockIdx.y;
    const int s = map[k];
    const int t = blockIdx.x * 1024 + threadIdx.x * 4;
    *(float4*)(out + (size_t)k * T_DIM + t) =
        *(const float4*)(proj + (size_t)s * T_DIM + t);
}

// ---------------------------------------------------------------------------
extern "C" void kernel_launch(void* const* d_in, const int* in_sizes, int n_in,
                              void* d_out, int out_size, void* d_ws, size_t ws_size,
                              hipStream_t stream) {
    const float* image  = (const float*)d_in[0]; // [1, 576, 1152]
    const float* Wproj  = (const float*)d_in[1]; // [4096, 1152]
    const float* bproj  = (const float*)d_in[2]; // [4096]
    const float* Wscore = (const float*)d_in[3]; // [1, 4096]
    const float* bscore = (const float*)d_in[4]; // [1]
    const float* pos    = (const float*)d_in[5]; // [576, 4096]
    float* out = (float*)d_out;                  // [1, 128, 4096]

    float* proj   = (float*)d_ws;                        // 576*4096 f32
    float* logits = proj + (size_t)S_DIM * T_DIM;        // 576 f32
    int*   map    = (int*)(logits + S_DIM);              // 128 i32

    dim3 g1(T_DIM / 128, S_DIM / 64);                    // 32 x 9 blocks
    proj_gemm_kernel<<<g1, 256, 0, stream>>>(image, Wproj, bproj, pos, proj);

    score_kernel<<<S_DIM, 256, 0, stream>>>(proj, Wscore, bscore, logits);

    topk_kernel<<<1, S_DIM, 0, stream>>>(logits, map);

    dim3 g4(T_DIM / 1024, TOPK);                         // 4 x 128 blocks
    gather_kernel<<<g4, 256, 0, stream>>>(proj, map, out);
}